// BEVDepthHead_39505109188938
// MI455X (gfx1250) — compile-verified
//
#include <hip/hip_runtime.h>

// ---------------- constants ----------------
#define BB    64
#define CC    10
#define HH    256
#define WW    256
#define HWSZ  (HH*WW)            // 65536
#define KTOP  500
#define NBIN  4096
#define CANDMAX   4096           // final per-batch candidate cap
#define CAND0MAX  49152          // pool per-batch candidate cap (48K)
#define SEGS  8                  // per-batch segments for pool passes
#define TILE  4096               // floats per staged tile (16KB)

// order-preserving float -> uint key (bigger key == bigger float)
__device__ __forceinline__ unsigned fkey(float f) {
  unsigned u = __float_as_uint(f);
  return (u & 0x80000000u) ? ~u : (u | 0x80000000u);
}
__device__ __forceinline__ float kval(unsigned k) {
  unsigned u = (k & 0x80000000u) ? (k ^ 0x80000000u) : ~k;
  return __uint_as_float(u);
}

// Per-lane async copy of 16B global -> LDS (generic LDS ptr low 32 bits = LDS
// byte address per the aperture layout).
__device__ __forceinline__ void async_tile_load(const float* g, float* lbuf) {
  int tid = threadIdx.x;
#pragma unroll
  for (int u = 0; u < 4; ++u) {
    int e4 = u * 256 + tid;  // float4 index in tile
    unsigned lds32 = (unsigned)(unsigned long long)(const void*)(lbuf + e4 * 4);
    unsigned long long ga = (unsigned long long)(const void*)(g + e4 * 4);
    asm volatile("global_load_async_to_lds_b128 %0, %1, off"
                 :: "v"(lds32), "v"(ga) : "memory");
  }
}
__device__ __forceinline__ void wait_async_le4() {
  asm volatile("s_wait_asynccnt 0x4" ::: "memory");
}
__device__ __forceinline__ void wait_async_0() {
  asm volatile("s_wait_asynccnt 0x0" ::: "memory");
}

// Wave-aggregated append: one atomic per wave32 instead of one per hit lane.
// Call with the whole wave converged (pred as data, not branch).
__device__ __forceinline__ void append_cand(bool pred, unsigned long long val,
                                            unsigned* cnt, unsigned long long* buf,
                                            unsigned cap) {
  unsigned m = (unsigned)__ballot(pred);     // wave32 mask
  unsigned tot = __popc(m);
  if (tot == 0u) return;
  unsigned lane = threadIdx.x & 31u;
  unsigned base = 0u;
  if (lane == 0u) base = atomicAdd(cnt, tot);
  base = (unsigned)__shfl((int)base, 0, 32);
  if (pred) {
    unsigned off = base + __popc(m & ((1u << lane) - 1u));
    if (off < cap) buf[off] = val;
  }
}

// ---------------- pass 0: zero scratch ----------------
__global__ void zero_k(unsigned* hists, unsigned* cnt0, unsigned* cnt1) {
  unsigned i = blockIdx.x * 256u + threadIdx.x;
  if (i < 2u * BB * NBIN) hists[i] = 0u;
  if (i < BB) { cnt0[i] = 0u; cnt1[i] = 0u; }
}

// ============ single full scan: local hist -> local cutoff -> compact ======
// Each block owns one (b,c) 256KB slab. Builds a slab-local 4096-bin histogram
// (async-LDS staged), picks the local bin where cum>=K (slab top-K superset),
// then re-streams the slab (L2-hot) appending everything >= that bin's lower
// bound into the per-batch pool. Pool provably contains the batch top-K.
__global__ __launch_bounds__(256)
void scan_select_k(const float* __restrict__ heat,
                   unsigned long long* __restrict__ cand0,
                   unsigned* __restrict__ cnt0) {
  __shared__ __align__(16) float tbuf[2][TILE];    // 32 KB
  __shared__ unsigned lhist[NBIN];                 // 16 KB
  __shared__ unsigned csum[256];
  __shared__ unsigned sklo;
  const int blk = blockIdx.x;
  const int b   = blk / CC;
  const int tid = threadIdx.x;
  const unsigned cbase = (unsigned)(blk % CC) * HWSZ;
  const float* slab = heat + (size_t)blk * HWSZ;
  unsigned long long* buf = cand0 + (size_t)b * CAND0MAX;
  unsigned* cnt = cnt0 + b;

  // ---- phase A: slab-local histogram, async double-buffered ----
  for (int i = tid; i < NBIN; i += 256) lhist[i] = 0u;
  async_tile_load(slab, tbuf[0]);
  __syncthreads();
  for (int t = 0; t < HWSZ / TILE; ++t) {
    if (t + 1 < HWSZ / TILE) {
      async_tile_load(slab + (size_t)(t + 1) * TILE, tbuf[(t + 1) & 1]);
      wait_async_le4();                  // tile t landed, t+1 in flight
    } else {
      wait_async_0();
    }
    const float4* tb = (const float4*)tbuf[t & 1];
#pragma unroll
    for (int u = 0; u < 4; ++u) {
      float4 v = tb[u * 256 + tid];      // same elements this lane async-loaded
      atomicAdd(&lhist[fkey(v.x) >> 20], 1u);
      atomicAdd(&lhist[fkey(v.y) >> 20], 1u);
      atomicAdd(&lhist[fkey(v.z) >> 20], 1u);
      atomicAdd(&lhist[fkey(v.w) >> 20], 1u);
    }
    __syncthreads();
  }

  // ---- phase B: block-local suffix select (first bin from top, cum>=K) ----
  {
    unsigned sum = 0;
#pragma unroll
    for (int i = 0; i < NBIN / 256; ++i) sum += lhist[tid * (NBIN / 256) + i];
    csum[tid] = sum;
    __syncthreads();
    if (tid == 0) {
      unsigned cum = 0; int chunk = 0;
      for (int c = 255; c >= 0; --c) {
        if (cum + csum[c] >= (unsigned)KTOP) { chunk = c; break; }
        cum += csum[c];
      }
      int bin = chunk * (NBIN / 256);
      for (int i = NBIN / 256 - 1; i >= 0; --i) {
        cum += lhist[chunk * (NBIN / 256) + i];
        if (cum >= (unsigned)KTOP) { bin = chunk * (NBIN / 256) + i; break; }
      }
      sklo = (unsigned)bin << 20;        // local cutoff (bin lower bound)
    }
    __syncthreads();
  }
  const unsigned klo = sklo;

  // ---- phase C: re-stream slab (L2-hot), append >= klo to batch pool ----
  async_tile_load(slab, tbuf[0]);
  __syncthreads();
  for (int t = 0; t < HWSZ / TILE; ++t) {
    if (t + 1 < HWSZ / TILE) {
      async_tile_load(slab + (size_t)(t + 1) * TILE, tbuf[(t + 1) & 1]);
      wait_async_le4();
    } else {
      wait_async_0();
    }
    const float4* tb = (const float4*)tbuf[t & 1];
#pragma unroll
    for (int u = 0; u < 4; ++u) {
      int e4 = u * 256 + tid;
      float4 v = tb[e4];
      unsigned idx0 = cbase + (unsigned)(t * TILE) + (unsigned)e4 * 4u;
      float arr[4] = {v.x, v.y, v.z, v.w};
#pragma unroll
      for (int j = 0; j < 4; ++j) {
        unsigned k = fkey(arr[j]);
        unsigned long long pv =
            ((unsigned long long)k << 32) | (unsigned long long)(0xFFFFFFFFu - (idx0 + j));
        append_cand(k >= klo, pv, cnt, buf, CAND0MAX);
      }
    }
    __syncthreads();
  }
}

// ---------------- parallel "first bin from top where cum >= K" -------------
__device__ __forceinline__ void suffix_select(const unsigned* __restrict__ h,
                                              unsigned start_cum,
                                              unsigned* bin_out, unsigned* above_out) {
  __shared__ unsigned csum[256];
  const int tid = threadIdx.x;
  unsigned sum = 0;
#pragma unroll
  for (int i = 0; i < NBIN / 256; ++i) sum += h[tid * (NBIN / 256) + i];
  csum[tid] = sum;
  __syncthreads();
  if (tid == 0) {
    unsigned cum = start_cum;
    int chunk = 0;
    for (int c = 255; c >= 0; --c) {
      if (cum + csum[c] >= (unsigned)KTOP) { chunk = c; break; }
      cum += csum[c];
    }
    unsigned prev = cum;
    int bin = chunk * (NBIN / 256);
    for (int i = NBIN / 256 - 1; i >= 0; --i) {
      unsigned c = h[chunk * (NBIN / 256) + i];
      prev = cum; cum += c;
      if (cum >= (unsigned)KTOP) { bin = chunk * (NBIN / 256) + i; break; }
    }
    *bin_out = (unsigned)bin;
    *above_out = prev;
  }
}

// ---------------- pool pass: coarse histogram (bits 31:20) -----------------
__global__ __launch_bounds__(256)
void pool_hist_coarse_k(const unsigned long long* __restrict__ cand0,
                        const unsigned* __restrict__ cnt0,
                        unsigned* __restrict__ hist1) {
  __shared__ unsigned lhist[NBIN];
  const int b = blockIdx.x, s = blockIdx.y, tid = threadIdx.x;
  const unsigned SEG = CAND0MAX / SEGS;
  const unsigned n = min(cnt0[b], (unsigned)CAND0MAX);
  const unsigned long long* buf = cand0 + (size_t)b * CAND0MAX;

  for (int i = tid; i < NBIN; i += 256) lhist[i] = 0u;
  __syncthreads();
  unsigned hi = min((unsigned)(s + 1) * SEG, n);
  for (unsigned i = (unsigned)s * SEG + tid; i < hi; i += 256)
    atomicAdd(&lhist[(unsigned)(buf[i] >> 52)], 1u);   // key>>20 == packed>>52
  __syncthreads();
  for (int i = tid; i < NBIN; i += 256) {
    unsigned c = lhist[i];
    if (c) atomicAdd(&hist1[(size_t)b * NBIN + i], c);
  }
}

__global__ __launch_bounds__(256)
void find_bin1_k(const unsigned* __restrict__ hist1,
                 unsigned* __restrict__ tc, unsigned* __restrict__ above) {
  suffix_select(hist1 + (size_t)blockIdx.x * NBIN, 0u,
                &tc[blockIdx.x], &above[blockIdx.x]);
}

// ---------------- pool pass: fine histogram (bits 19:8 inside tc) ----------
__global__ __launch_bounds__(256)
void pool_hist_fine_k(const unsigned long long* __restrict__ cand0,
                      const unsigned* __restrict__ cnt0,
                      const unsigned* __restrict__ tc,
                      unsigned* __restrict__ hist2) {
  __shared__ unsigned lhist[NBIN];
  const int b = blockIdx.x, s = blockIdx.y, tid = threadIdx.x;
  const unsigned SEG = CAND0MAX / SEGS;
  const unsigned n = min(cnt0[b], (unsigned)CAND0MAX);
  const unsigned tcb = tc[b];
  const unsigned long long* buf = cand0 + (size_t)b * CAND0MAX;

  for (int i = tid; i < NBIN; i += 256) lhist[i] = 0u;
  __syncthreads();
  unsigned hi = min((unsigned)(s + 1) * SEG, n);
  for (unsigned i = (unsigned)s * SEG + tid; i < hi; i += 256) {
    unsigned key = (unsigned)(buf[i] >> 32);
    if ((key >> 20) == tcb) atomicAdd(&lhist[(key >> 8) & 0xFFFu], 1u);
  }
  __syncthreads();
  for (int i = tid; i < NBIN; i += 256) {
    unsigned c = lhist[i];
    if (c) atomicAdd(&hist2[(size_t)b * NBIN + i], c);
  }
}

__global__ __launch_bounds__(256)
void find_bin2_k(const unsigned* __restrict__ hist2,
                 const unsigned* __restrict__ tc,
                 const unsigned* __restrict__ above,
                 unsigned* __restrict__ keylo) {
  __shared__ unsigned sbin, sabove;
  suffix_select(hist2 + (size_t)blockIdx.x * NBIN, above[blockIdx.x], &sbin, &sabove);
  __syncthreads();
  if (threadIdx.x == 0) keylo[blockIdx.x] = (tc[blockIdx.x] << 20) | (sbin << 8);
}

// ---------------- pool pass: final compaction cand0 -> cand1 ---------------
__global__ __launch_bounds__(256)
void compact_fine_k(const unsigned long long* __restrict__ cand0,
                    const unsigned* __restrict__ cnt0,
                    const unsigned* __restrict__ keylo,
                    unsigned long long* __restrict__ cand1,
                    unsigned* __restrict__ cnt1) {
  const int b = blockIdx.x, s = blockIdx.y, tid = threadIdx.x;
  const unsigned SEG = CAND0MAX / SEGS;
  const unsigned n = min(cnt0[b], (unsigned)CAND0MAX);
  const unsigned long long thr = (unsigned long long)keylo[b] << 32;
  const unsigned long long* buf = cand0 + (size_t)b * CAND0MAX;
  unsigned long long* obuf = cand1 + (size_t)b * CANDMAX;
  unsigned* cnt = cnt1 + b;

  // uniform trip count (wave stays converged for aggregated append)
  for (unsigned i0 = (unsigned)s * SEG; i0 < (unsigned)(s + 1) * SEG; i0 += 256) {
    unsigned i = i0 + tid;
    unsigned long long pv = buf[i];        // in-bounds of scratch; use guarded by pred
    bool pred = (i < n) && (pv >= thr);
    append_cand(pred, pv, cnt, obuf, CANDMAX);
  }
}

// ---------------- final: sort candidates, gather & decode ------------------
__global__ __launch_bounds__(1024)
void decode_k(const float* __restrict__ rot_sine, const float* __restrict__ rot_cosine,
              const float* __restrict__ hei, const float* __restrict__ dim,
              const float* __restrict__ vel, const float* __restrict__ reg,
              const unsigned long long* __restrict__ cand1,
              const unsigned* __restrict__ cnt1,
              float* __restrict__ out) {
  __shared__ unsigned long long s[CANDMAX];   // 32 KB
  const int b = blockIdx.x;
  const int tid = threadIdx.x;
  const unsigned n = min(cnt1[b], (unsigned)CANDMAX);

  for (int i = tid; i < CANDMAX; i += 1024)
    s[i] = (i < (int)n) ? cand1[(size_t)b * CANDMAX + i] : 0ull;
  __syncthreads();

  // bitonic sort, descending (value desc, index asc via ~idx packing)
  for (unsigned k = 2; k <= CANDMAX; k <<= 1) {
    for (unsigned j = k >> 1; j > 0; j >>= 1) {
      for (unsigned i = tid; i < CANDMAX; i += 1024) {
        unsigned ixj = i ^ j;
        if (ixj > i) {
          unsigned long long a = s[i], c = s[ixj];
          bool desc = ((i & k) == 0);
          if (desc ? (a < c) : (a > c)) { s[i] = c; s[ixj] = a; }
        }
      }
      __syncthreads();
    }
  }

  if (tid < KTOP) {
    unsigned long long p = s[tid];
    unsigned key = (unsigned)(p >> 32);
    unsigned idx = 0xFFFFFFFFu - (unsigned)(p & 0xFFFFFFFFull);
    float score;
    if (p == 0ull) { idx = 0u; score = -1.0e30f; }
    else           { score = kval(key); }
    unsigned cls = idx / HWSZ;
    unsigned hw  = idx % HWSZ;
    float xi = (float)(hw % WW);
    float yi = (float)(hw / WW);

    size_t o1 = (size_t)b * HWSZ + hw;                       // [B,1,H,W]
    float r0 = reg[(size_t)(b * 2 + 0) * HWSZ + hw];
    float r1 = reg[(size_t)(b * 2 + 1) * HWSZ + hw];
    float rs = rot_sine[o1];
    float rc = rot_cosine[o1];
    float hz = hei[o1];
    float d0 = dim[(size_t)(b * 3 + 0) * HWSZ + hw];
    float d1 = dim[(size_t)(b * 3 + 1) * HWSZ + hw];
    float d2 = dim[(size_t)(b * 3 + 2) * HWSZ + hw];
    float v0 = vel[(size_t)(b * 2 + 0) * HWSZ + hw];
    float v1 = vel[(size_t)(b * 2 + 1) * HWSZ + hw];

    float x = (xi + r0) * 0.8f - 51.2f;   // OUT_SIZE_FACTOR*VOXEL + PC_RANGE
    float y = (yi + r1) * 0.8f - 51.2f;
    float rot = atan2f(rs, rc);

    size_t bk = (size_t)b * KTOP + tid;
    float* box = out + bk * 9;
    box[0] = x;  box[1] = y;  box[2] = hz;
    box[3] = d0; box[4] = d1; box[5] = d2;
    box[6] = rot; box[7] = v0; box[8] = v1;

    const size_t BK = (size_t)BB * KTOP;
    out[BK * 9 + bk] = score;                       // final_scores
    out[BK * 10 + bk] = (float)cls;                 // final_preds
    bool keep = (score > 0.1f) &&
                (x >= -61.2f) && (x <= 61.2f) &&
                (y >= -61.2f) && (y <= 61.2f) &&
                (hz >= -10.0f) && (hz <= 10.0f);
    out[BK * 11 + bk] = keep ? 1.0f : 0.0f;         // keep mask
  }
}

extern "C" void kernel_launch(void* const* d_in, const int* in_sizes, int n_in,
                              void* d_out, int out_size, void* d_ws, size_t ws_size,
                              hipStream_t stream) {
  const float* heat       = (const float*)d_in[0];
  const float* rot_sine   = (const float*)d_in[1];
  const float* rot_cosine = (const float*)d_in[2];
  const float* hei        = (const float*)d_in[3];
  const float* dim        = (const float*)d_in[4];
  const float* vel        = (const float*)d_in[5];
  const float* reg        = (const float*)d_in[6];
  float* out = (float*)d_out;

  char* ws = (char*)d_ws;
  unsigned* hist1 = (unsigned*)ws;                                  // 1 MB
  unsigned* hist2 = (unsigned*)(ws + (1u << 20));                   // 1 MB (contiguous)
  unsigned* tc    = (unsigned*)(ws + (2u << 20));
  unsigned* above = (unsigned*)(ws + (2u << 20) + 1024);
  unsigned* keylo = (unsigned*)(ws + (2u << 20) + 2048);
  unsigned* cnt0  = (unsigned*)(ws + (2u << 20) + 3072);
  unsigned* cnt1  = (unsigned*)(ws + (2u << 20) + 4096);
  unsigned long long* cand1 = (unsigned long long*)(ws + (2u << 20) + 8192);  // 2 MB
  unsigned long long* cand0 = (unsigned long long*)(ws + (6u << 20));         // 24 MB

  zero_k<<<2048, 256, 0, stream>>>(hist1, cnt0, cnt1);
  scan_select_k<<<BB * CC, 256, 0, stream>>>(heat, cand0, cnt0);    // the one full scan
  pool_hist_coarse_k<<<dim3(BB, SEGS), 256, 0, stream>>>(cand0, cnt0, hist1);
  find_bin1_k<<<BB, 256, 0, stream>>>(hist1, tc, above);
  pool_hist_fine_k<<<dim3(BB, SEGS), 256, 0, stream>>>(cand0, cnt0, tc, hist2);
  find_bin2_k<<<BB, 256, 0, stream>>>(hist2, tc, above, keylo);
  compact_fine_k<<<dim3(BB, SEGS), 256, 0, stream>>>(cand0, cnt0, keylo, cand1, cnt1);
  decode_k<<<BB, 1024, 0, stream>>>(rot_sine, rot_cosine, hei, dim, vel, reg,
                                    cand1, cnt1, out);
}